// LatentReciprocalLongRange_66881230733963
// MI455X (gfx1250) — compile-verified
//
#include <hip/hip_runtime.h>
#include <math.h>

typedef float v2f __attribute__((ext_vector_type(2)));
typedef float v8f __attribute__((ext_vector_type(8)));

#define NATOMS 400000
#define FDIM 128
#define HDIM 128
#define W1PITCH 130            // transposed-W1 LDS pitch: 2-bank lane stride, 8B aligned
#define MESH 64
#define MESH3 (MESH * MESH * MESH)
#define KFLOOR 1e-6f
#define PI_F 3.14159265358979323846f

// ---------------------------------------------------------------------------
// Prep: 3x3 inverse of cell + |det| (one thread; consumed by spread/spectral)
// ---------------------------------------------------------------------------
__global__ void prep_kernel(const float* __restrict__ cell, float* __restrict__ prep) {
    if (threadIdx.x == 0 && blockIdx.x == 0) {
        float c[9];
        #pragma unroll
        for (int i = 0; i < 9; ++i) c[i] = cell[i];
        float det = c[0]*(c[4]*c[8]-c[5]*c[7])
                  - c[1]*(c[3]*c[8]-c[5]*c[6])
                  + c[2]*(c[3]*c[7]-c[4]*c[6]);
        float id = 1.0f / det;
        float inv[9];
        inv[0] = (c[4]*c[8]-c[5]*c[7])*id;
        inv[1] = (c[2]*c[7]-c[1]*c[8])*id;
        inv[2] = (c[1]*c[5]-c[2]*c[4])*id;
        inv[3] = (c[5]*c[6]-c[3]*c[8])*id;
        inv[4] = (c[0]*c[8]-c[2]*c[6])*id;
        inv[5] = (c[2]*c[3]-c[0]*c[5])*id;
        inv[6] = (c[3]*c[7]-c[4]*c[6])*id;
        inv[7] = (c[1]*c[6]-c[0]*c[7])*id;
        inv[8] = (c[0]*c[4]-c[1]*c[3])*id;
        #pragma unroll
        for (int i = 0; i < 9; ++i) prep[i] = inv[i];
        prep[9] = fmaxf(fabsf(det), KFLOOR);
    }
}

// ---------------------------------------------------------------------------
// Fused MLP:  src = silu(X @ W1 + b1) @ w2 + b2   via V_WMMA_F32_16X16X4_F32
// Block = 256 threads = 8 waves; wave handles 16 rows; grid = N/128 = 3125.
// W1 is staged TRANSPOSED in LDS so each B fragment is one aligned ds_load_b64.
// ---------------------------------------------------------------------------
__global__ __launch_bounds__(256) void mlp_wmma_kernel(
    const float* __restrict__ X, const float* __restrict__ W1,
    const float* __restrict__ B1, const float* __restrict__ W2,
    const float* __restrict__ B2, float* __restrict__ src)
{
    __shared__ float lds_w1t[HDIM * W1PITCH];   // 65 KB, w1t[n][k] = W1[k][n]

    const int tid = threadIdx.x;
    // stage transposed: coalesced global read, bank-spread LDS write
    for (int i = tid; i < FDIM * HDIM; i += 256) {
        const int k = i >> 7;       // W1 row   (K index)
        const int n = i & 127;      // W1 col   (hidden index)
        lds_w1t[n * W1PITCH + k] = W1[i];
    }
    __syncthreads();

    const int wave = tid >> 5;
    const int lane = tid & 31;
    const int half = lane >> 4;     // K-half selector for A/B fragments
    const int l16  = lane & 15;
    const int row0 = blockIdx.x * 128 + wave * 16;
    const int arow = row0 + l16;    // A rows: M = l16 in both lane halves

    v8f acc[8];
    #pragma unroll
    for (int nt = 0; nt < 8; ++nt)
        acc[nt] = (v8f){0.f,0.f,0.f,0.f,0.f,0.f,0.f,0.f};

    const float* aptr = X + (size_t)arow * FDIM + half * 2;
    const float* bbase = lds_w1t + l16 * W1PITCH + half * 2;

    // K = 128 in 32 steps of 4; 8 N-tiles of 16 hidden cols kept live.
    for (int kk = 0; kk < 32; ++kk) {
        // A 16x4 f32 fragment: lanes 0-15 K={0,1}, lanes 16-31 K={2,3}
        const v2f a = *(const v2f*)(aptr + kk * 4);
        #pragma unroll
        for (int nt = 0; nt < 8; ++nt) {
            // contiguous pair {k, k+1} from transposed LDS -> single ds_load_b64
            const v2f b = *(const v2f*)(bbase + nt * 16 * W1PITCH + kk * 4);
            acc[nt] = __builtin_amdgcn_wmma_f32_16x16x4_f32(
                false, a, false, b, (short)0, acc[nt], false, false);
        }
    }

    // Epilogue: bias + SiLU + dot with w2 (C==1), fused per lane.
    // C/D layout: element v of lane -> (M = v + 8*half, N = l16).
    float partial[8];
    #pragma unroll
    for (int v = 0; v < 8; ++v) partial[v] = 0.f;

    #pragma unroll
    for (int nt = 0; nt < 8; ++nt) {
        const int hcol = nt * 16 + l16;
        const float bb = B1[hcol];
        const float ww = W2[hcol];
        #pragma unroll
        for (int v = 0; v < 8; ++v) {
            float x = acc[nt][v] + bb;
            float s = x / (1.0f + __expf(-x));   // SiLU
            partial[v] += s * ww;
        }
    }
    // reduce across the 16 lanes of each half (butterfly stays inside halves)
    #pragma unroll
    for (int v = 0; v < 8; ++v) {
        float p = partial[v];
        p += __shfl_xor(p, 1, 32);
        p += __shfl_xor(p, 2, 32);
        p += __shfl_xor(p, 4, 32);
        p += __shfl_xor(p, 8, 32);
        partial[v] = p;
    }
    const float b2v = B2[0];
    if (l16 == 0) {
        #pragma unroll
        for (int v = 0; v < 8; ++v)
            src[row0 + half * 8 + v] = partial[v] + b2v;
    }
}

// ---------------------------------------------------------------------------
// Zero helper
// ---------------------------------------------------------------------------
__global__ void zero_kernel(float* __restrict__ p, int n) {
    int i = blockIdx.x * blockDim.x + threadIdx.x;
    if (i < n) p[i] = 0.f;
}

// ---------------------------------------------------------------------------
// CIC helpers
// ---------------------------------------------------------------------------
__device__ __forceinline__ void cic_setup(const float* __restrict__ pos,
                                          const float* __restrict__ prep,
                                          int n, int base[3], float fo[3])
{
    float px = pos[n * 3 + 0], py = pos[n * 3 + 1], pz = pos[n * 3 + 2];
    float inv[9];
    #pragma unroll
    for (int i = 0; i < 9; ++i) inv[i] = prep[i];
    float f[3];
    f[0] = px * inv[0] + py * inv[3] + pz * inv[6];
    f[1] = px * inv[1] + py * inv[4] + pz * inv[7];
    f[2] = px * inv[2] + py * inv[5] + pz * inv[8];
    #pragma unroll
    for (int d = 0; d < 3; ++d) {
        float fr = f[d] - floorf(f[d]);           // wrap to [0,1)
        float sc = fr * (float)MESH;
        float fb = floorf(sc);
        fo[d] = sc - fb;
        int bi = (int)fb;
        bi = ((bi % MESH) + MESH) % MESH;         // guard fr==1.0 rounding
        base[d] = bi;
    }
}

__global__ void spread_kernel(const float* __restrict__ pos,
                              const float* __restrict__ src,
                              const float* __restrict__ prep,
                              float* __restrict__ mesh)
{
    int n = blockIdx.x * blockDim.x + threadIdx.x;
    if (n >= NATOMS) return;
    int base[3]; float fo[3];
    cic_setup(pos, prep, n, base, fo);
    float sv = src[n];
    #pragma unroll
    for (int c = 0; c < 8; ++c) {
        int bx = (c >> 2) & 1, by = (c >> 1) & 1, bz = c & 1;
        float w = (bx ? fo[0] : 1.f - fo[0])
                * (by ? fo[1] : 1.f - fo[1])
                * (bz ? fo[2] : 1.f - fo[2]);
        int ix = base[0] + bx; if (ix >= MESH) ix -= MESH;
        int iy = base[1] + by; if (iy >= MESH) iy -= MESH;
        int iz = base[2] + bz; if (iz >= MESH) iz -= MESH;
        atomicAdd(&mesh[(ix * MESH + iy) * MESH + iz], sv * w);
    }
}

// ---------------------------------------------------------------------------
// Batched 64-point radix-2 FFT pass (DIT, bit-reversed input), LDS-based.
// 4 lines per 256-thread block; 4096 lines per axis.
// line base = (l / A)*B + (l % A)*Cc ; element g = base + t*stride
// ---------------------------------------------------------------------------
__device__ __forceinline__ int bitrev6(int x) {
    int r = 0;
    #pragma unroll
    for (int i = 0; i < 6; ++i) r = (r << 1) | ((x >> i) & 1);
    return r;
}

__global__ __launch_bounds__(256) void fft_pass_kernel(
    float2* __restrict__ data, const float* __restrict__ realIn,
    float* __restrict__ realOut,
    int stride, int A, int B, int Cc, float dirSign, float scale)
{
    __shared__ float2 buf[4][65];
    const int tid   = threadIdx.x;
    const int lline = tid >> 6;          // 0..3
    const int t     = tid & 63;
    const int gl    = blockIdx.x * 4 + lline;
    const int base  = (gl / A) * B + (gl % A) * Cc;
    const int g     = base + t * stride;

    float2 val;
    if (realIn) { val.x = realIn[g]; val.y = 0.f; }
    else        { val = data[g]; }
    buf[lline][bitrev6(t)] = val;
    __syncthreads();

    #pragma unroll
    for (int s = 1; s <= 6; ++s) {
        const int m = 1 << s, h = m >> 1;
        if (t < 32) {
            int j   = t & (h - 1);
            int grp = t >> (s - 1);
            int i0  = grp * m + j;
            int i1  = i0 + h;
            float ang = dirSign * 2.0f * PI_F * (float)j / (float)m;
            float sw, cw;
            __sincosf(ang, &sw, &cw);
            float2 u = buf[lline][i0];
            float2 b = buf[lline][i1];
            float2 v;
            v.x = cw * b.x - sw * b.y;
            v.y = cw * b.y + sw * b.x;
            buf[lline][i0] = make_float2(u.x + v.x, u.y + v.y);
            buf[lline][i1] = make_float2(u.x - v.x, u.y - v.y);
        }
        __syncthreads();
    }

    float2 o = buf[lline][t];
    o.x *= scale; o.y *= scale;
    if (realOut) realOut[g] = o.x;
    else         data[g] = o;
}

// ---------------------------------------------------------------------------
// Green's function multiply in k-space
// ---------------------------------------------------------------------------
__global__ void spectral_kernel(float2* __restrict__ data, const float* __restrict__ prep)
{
    int p = blockIdx.x * blockDim.x + threadIdx.x;
    if (p >= MESH3) return;
    int iz = p & 63, iy = (p >> 6) & 63, ix = p >> 12;
    float fx = (float)(ix < 32 ? ix : ix - 64);
    float fy = (float)(iy < 32 ? iy : iy - 64);
    float fz = (float)(iz < 32 ? iz : iz - 64);
    float inv[9];
    #pragma unroll
    for (int i = 0; i < 9; ++i) inv[i] = prep[i];
    float tp = 2.0f * PI_F;
    float kx = tp * (fx * inv[0] + fy * inv[3] + fz * inv[6]);
    float ky = tp * (fx * inv[1] + fy * inv[4] + fz * inv[7]);
    float kz = tp * (fx * inv[2] + fy * inv[5] + fz * inv[8]);
    float kn = sqrtf(kx * kx + ky * ky + kz * kz);
    float vol = prep[9];
    float sk = fmaxf(kn, KFLOOR);
    float spec = (kn > KFLOOR) ? (4.0f * PI_F) / (sk * sk) / vol : 0.f;
    float2 v = data[p];
    v.x *= spec; v.y *= spec;
    data[p] = v;
}

// ---------------------------------------------------------------------------
// CIC gather + per-atom energy
// ---------------------------------------------------------------------------
__global__ void gather_kernel(const float* __restrict__ pos,
                              const float* __restrict__ src,
                              const float* __restrict__ prep,
                              const float* __restrict__ filt,
                              float* __restrict__ out)
{
    int n = blockIdx.x * blockDim.x + threadIdx.x;
    if (n >= NATOMS) return;
    int base[3]; float fo[3];
    cic_setup(pos, prep, n, base, fo);
    float pot = 0.f;
    #pragma unroll
    for (int c = 0; c < 8; ++c) {
        int bx = (c >> 2) & 1, by = (c >> 1) & 1, bz = c & 1;
        float w = (bx ? fo[0] : 1.f - fo[0])
                * (by ? fo[1] : 1.f - fo[1])
                * (bz ? fo[2] : 1.f - fo[2]);
        int ix = base[0] + bx; if (ix >= MESH) ix -= MESH;
        int iy = base[1] + by; if (iy >= MESH) iy -= MESH;
        int iz = base[2] + bz; if (iz >= MESH) iz -= MESH;
        pot += w * filt[(ix * MESH + iy) * MESH + iz];
    }
    out[n] = 0.5f * src[n] * pot;
}

// ---------------------------------------------------------------------------
// Launch
// ---------------------------------------------------------------------------
extern "C" void kernel_launch(void* const* d_in, const int* in_sizes, int n_in,
                              void* d_out, int out_size, void* d_ws, size_t ws_size,
                              hipStream_t stream) {
    const float* features = (const float*)d_in[0];
    const float* pos      = (const float*)d_in[1];
    // d_in[2] = batch (all zeros, unused)
    const float* cell     = (const float*)d_in[3];
    const float* w1       = (const float*)d_in[4];
    const float* b1       = (const float*)d_in[5];
    const float* w2       = (const float*)d_in[6];
    const float* b2       = (const float*)d_in[7];
    float* out = (float*)d_out;

    char* ws = (char*)d_ws;
    float*  prep = (float*)(ws + 0);                 // 10 floats
    float*  src  = (float*)(ws + 256);               // 400000 floats (1.6 MB)
    float*  mesh = (float*)(ws + 1600256);           // 262144 floats (1 MB)
    float2* cplx = (float2*)(ws + 2648832);          // 262144 float2 (2 MB)
    float*  filt = (float*)(ws + 4745984);           // 262144 floats (1 MB)

    prep_kernel<<<1, 64, 0, stream>>>(cell, prep);

    mlp_wmma_kernel<<<NATOMS / 128, 256, 0, stream>>>(features, w1, b1, w2, b2, src);

    zero_kernel<<<(MESH3 + 255) / 256, 256, 0, stream>>>(mesh, MESH3);
    spread_kernel<<<(NATOMS + 255) / 256, 256, 0, stream>>>(pos, src, prep, mesh);

    // forward FFT: z (stride 1), y (stride 64), x (stride 4096)
    fft_pass_kernel<<<1024, 256, 0, stream>>>(cplx, mesh, nullptr, 1,    1,  64,   0, -1.f, 1.f);
    fft_pass_kernel<<<1024, 256, 0, stream>>>(cplx, nullptr, nullptr, 64, 64, 4096, 1, -1.f, 1.f);
    fft_pass_kernel<<<1024, 256, 0, stream>>>(cplx, nullptr, nullptr, 4096, 1, 1,   0, -1.f, 1.f);

    spectral_kernel<<<(MESH3 + 255) / 256, 256, 0, stream>>>(cplx, prep);

    // inverse FFT (scale 1/64 per pass); last pass writes real part
    fft_pass_kernel<<<1024, 256, 0, stream>>>(cplx, nullptr, nullptr, 4096, 1, 1,   0, +1.f, 1.f / 64.f);
    fft_pass_kernel<<<1024, 256, 0, stream>>>(cplx, nullptr, nullptr, 64, 64, 4096, 1, +1.f, 1.f / 64.f);
    fft_pass_kernel<<<1024, 256, 0, stream>>>(cplx, nullptr, filt,   1,    1,  64,  0, +1.f, 1.f / 64.f);

    gather_kernel<<<(NATOMS + 255) / 256, 256, 0, stream>>>(pos, src, prep, filt, out);
}